// UnifiedEstimatorQNN_65481071400813
// MI455X (gfx1250) — compile-verified
//
#include <hip/hip_runtime.h>

typedef __attribute__((ext_vector_type(2))) float v2f;
typedef __attribute__((ext_vector_type(8))) float v8f;

#define NW    12
#define DIM   4096
#define BATCH 512

struct PermMasks { int m[2][NW]; };   // per-layer GF(2) column masks: src(j)=XOR_{b in j} m[b]

static __device__ __forceinline__ v2f mk2(float a, float b) {
  v2f r; r.x = a; r.y = b; return r;
}

static __device__ __forceinline__ v8f wmma4(v2f a, v2f b, v8f c) {
  // D(16x16,f32) = A(16x4,f32) * B(4x16,f32) + C
  return __builtin_amdgcn_wmma_f32_16x16x4_f32(false, a, false, b, (short)0, c,
                                               false, false);
}

// ---- quartet gate on bits 11..8 (wires 0..3): D[c][r] = sum_j U[c][j] S[j][r]
static __device__ __forceinline__ void apply_q0(const float* Ur, const float* Ui,
                                                float* sr, float* si,
                                                int lane, int wid) {
  const int M = lane & 15;       // A row / B,D col
  const int half = lane >> 4;
  v2f aR[4], aI[4], aN[4];
#pragma unroll
  for (int k = 0; k < 4; ++k) {
    int kk = k * 4 + half * 2;
    aR[k] = mk2(Ur[M * 16 + kk], Ur[M * 16 + kk + 1]);
    aI[k] = mk2(Ui[M * 16 + kk], Ui[M * 16 + kk + 1]);
    aN[k] = -aI[k];
  }
#pragma unroll
  for (int t = 0; t < 2; ++t) {
    int cb = (wid * 2 + t) * 16;   // column tile base (low 8 bits of index)
    v2f bR[4], bI[4];
#pragma unroll
    for (int k = 0; k < 4; ++k) {
      int kk = k * 4 + half * 2;
      bR[k] = mk2(sr[kk * 256 + cb + M], sr[(kk + 1) * 256 + cb + M]);
      bI[k] = mk2(si[kk * 256 + cb + M], si[(kk + 1) * 256 + cb + M]);
    }
    v8f dr = {0,0,0,0,0,0,0,0}, di = {0,0,0,0,0,0,0,0};
#pragma unroll
    for (int k = 0; k < 4; ++k) dr = wmma4(aR[k], bR[k], dr);
#pragma unroll
    for (int k = 0; k < 4; ++k) dr = wmma4(aN[k], bI[k], dr);
#pragma unroll
    for (int k = 0; k < 4; ++k) di = wmma4(aR[k], bI[k], di);
#pragma unroll
    for (int k = 0; k < 4; ++k) di = wmma4(aI[k], bR[k], di);
#pragma unroll
    for (int m = 0; m < 8; ++m) {
      int row = m + half * 8;
      sr[row * 256 + cb + M] = dr[m];
      si[row * 256 + cb + M] = di[m];
    }
  }
}

// ---- quartet gate on bits 7..4 (wires 4..7): per slice a: D_a = U @ S_a
static __device__ __forceinline__ void apply_q1(const float* Ur, const float* Ui,
                                                float* sr, float* si,
                                                int lane, int wid) {
  const int M = lane & 15;
  const int half = lane >> 4;
  v2f aR[4], aI[4], aN[4];
#pragma unroll
  for (int k = 0; k < 4; ++k) {
    int kk = k * 4 + half * 2;
    aR[k] = mk2(Ur[M * 16 + kk], Ur[M * 16 + kk + 1]);
    aI[k] = mk2(Ui[M * 16 + kk], Ui[M * 16 + kk + 1]);
    aN[k] = -aI[k];
  }
#pragma unroll
  for (int t = 0; t < 2; ++t) {
    int base = (wid * 2 + t) * 256;   // slice over bits 11..8
    v2f bR[4], bI[4];
#pragma unroll
    for (int k = 0; k < 4; ++k) {
      int kk = k * 4 + half * 2;
      bR[k] = mk2(sr[base + kk * 16 + M], sr[base + (kk + 1) * 16 + M]);
      bI[k] = mk2(si[base + kk * 16 + M], si[base + (kk + 1) * 16 + M]);
    }
    v8f dr = {0,0,0,0,0,0,0,0}, di = {0,0,0,0,0,0,0,0};
#pragma unroll
    for (int k = 0; k < 4; ++k) dr = wmma4(aR[k], bR[k], dr);
#pragma unroll
    for (int k = 0; k < 4; ++k) dr = wmma4(aN[k], bI[k], dr);
#pragma unroll
    for (int k = 0; k < 4; ++k) di = wmma4(aR[k], bI[k], di);
#pragma unroll
    for (int k = 0; k < 4; ++k) di = wmma4(aI[k], bR[k], di);
#pragma unroll
    for (int m = 0; m < 8; ++m) {
      int row = m + half * 8;
      sr[base + row * 16 + M] = dr[m];
      si[base + row * 16 + M] = di[m];
    }
  }
}

// ---- quartet gate on bits 3..0 (wires 8..11): D = S @ U^T  (A = state rows)
static __device__ __forceinline__ void apply_q2(const float* Ur, const float* Ui,
                                                float* sr, float* si,
                                                int lane, int wid) {
  const int M = lane & 15;
  const int half = lane >> 4;
  v2f bR[4], bI[4], bN[4];          // B = U^T : b[k].p = U[N][kk+p], N == M
#pragma unroll
  for (int k = 0; k < 4; ++k) {
    int kk = k * 4 + half * 2;
    bR[k] = mk2(Ur[M * 16 + kk], Ur[M * 16 + kk + 1]);
    bI[k] = mk2(Ui[M * 16 + kk], Ui[M * 16 + kk + 1]);
    bN[k] = -bI[k];
  }
#pragma unroll
  for (int t = 0; t < 2; ++t) {
    int rb = (wid * 2 + t) * 16;    // row tile over bits 11..4
    v2f aR[4], aI[4];
#pragma unroll
    for (int k = 0; k < 4; ++k) {
      int kk = k * 4 + half * 2;
      aR[k] = mk2(sr[(rb + M) * 16 + kk], sr[(rb + M) * 16 + kk + 1]);
      aI[k] = mk2(si[(rb + M) * 16 + kk], si[(rb + M) * 16 + kk + 1]);
    }
    v8f dr = {0,0,0,0,0,0,0,0}, di = {0,0,0,0,0,0,0,0};
#pragma unroll
    for (int k = 0; k < 4; ++k) dr = wmma4(aR[k], bR[k], dr);
#pragma unroll
    for (int k = 0; k < 4; ++k) dr = wmma4(aI[k], bN[k], dr);
#pragma unroll
    for (int k = 0; k < 4; ++k) di = wmma4(aR[k], bI[k], di);
#pragma unroll
    for (int k = 0; k < 4; ++k) di = wmma4(aI[k], bR[k], di);
#pragma unroll
    for (int m = 0; m < 8; ++m) {
      int row = rb + m + half * 8;
      sr[row * 16 + M] = dr[m];
      si[row * 16 + M] = di[m];
    }
  }
}

// ---- fused CNOT-ring entangler: new[j] = old[ XOR_{b in j} colm[b] ]
// Gather to registers, barrier, scatter in place. Masks are uniform (SGPRs).
static __device__ __forceinline__ void permute(float* sr, float* si,
                                               const int* colm, int tid) {
  float vr[16], vi[16];
  int base = 0;
#pragma unroll
  for (int b = 0; b < 8; ++b)          // runtime tid bits -> v_xor with SGPR
    base ^= ((tid >> b) & 1) ? colm[b] : 0;
#pragma unroll
  for (int q = 0; q < 16; ++q) {       // compile-time q bits -> folded s_xor
    int hp = 0;
#pragma unroll
    for (int b = 0; b < 4; ++b)
      if ((q >> b) & 1) hp ^= colm[8 + b];
    int src = base ^ hp;
    vr[q] = sr[src];
    vi[q] = si[src];
  }
  __syncthreads();
#pragma unroll
  for (int q = 0; q < 16; ++q) {
    int j = tid + q * 256;
    sr[j] = vr[q];
    si[j] = vi[q];
  }
  __syncthreads();
}

// ---- prep: fuse 4 Rot gates per (layer, quartet) into 16x16 complex unitary
__global__ __launch_bounds__(256) void qnn_prep(const float* __restrict__ var,
                                                float* __restrict__ mats) {
  const int bl = blockIdx.x;          // 0..5 : layer*3 + quartet
  const int l = bl / 3, qq = bl % 3;
  const int t = threadIdx.x;
  const int i = t >> 4, j = t & 15;
  float re = 1.0f, im = 0.0f;
#pragma unroll
  for (int k = 0; k < 4; ++k) {
    int w = qq * 4 + k;
    const float* vp = var + (l * NW + w) * 3;
    float phi = vp[0], th = vp[1], om = vp[2];
    float c = cosf(0.5f * th), s = sinf(0.5f * th);
    float a  = 0.5f * (phi + om);
    float bb = 0.5f * (phi - om);
    int bi = (i >> (3 - k)) & 1, bj = (j >> (3 - k)) & 1;
    float er, ei;
    if (!bi && !bj)      { er =  cosf(a)  * c; ei = -sinf(a)  * c; }  // ep*c
    else if (!bi &&  bj) { er = -cosf(bb) * s; ei = -sinf(bb) * s; }  // -conj(em)*s
    else if ( bi && !bj) { er =  cosf(bb) * s; ei = -sinf(bb) * s; }  // em*s
    else                 { er =  cosf(a)  * c; ei =  sinf(a)  * c; }  // conj(ep)*c
    float nr = re * er - im * ei;
    float ni = re * ei + im * er;
    re = nr; im = ni;
  }
  mats[bl * 512 + t]       = re;
  mats[bl * 512 + 256 + t] = im;
}

// ---- main: one workgroup per batch sample, state resident in LDS
__global__ __launch_bounds__(256) void qnn_sim(const float* __restrict__ x,
                                               const float* __restrict__ mats_g,
                                               const float* __restrict__ head_w,
                                               const float* __restrict__ head_b,
                                               float* __restrict__ out,
                                               PermMasks pm) {
  __shared__ float sr[DIM];
  __shared__ float si[DIM];
  __shared__ float mats[6 * 512];
  __shared__ float trig[2 * NW];
  __shared__ float red[256];

  const int tid  = threadIdx.x;
  const int lane = tid & 31;
  const int wid  = tid >> 5;
  const int b    = blockIdx.x;

  for (int i2 = tid; i2 < 6 * 512; i2 += 256) mats[i2] = mats_g[i2];
  if (tid < NW) {
    float xv = x[b * NW + tid];
    trig[tid]      = cosf(0.5f * xv);
    trig[tid + NW] = sinf(0.5f * xv);
  }
  __syncthreads();

  // RY-encoded product state: amp[i] = prod_w (bit(11-w)? sin : cos)
  for (int idx = tid; idx < DIM; idx += 256) {
    float a = 1.0f;
#pragma unroll
    for (int w = 0; w < NW; ++w) {
      int bit = (idx >> (11 - w)) & 1;
      a *= bit ? trig[NW + w] : trig[w];
    }
    sr[idx] = a;
    si[idx] = 0.0f;
  }
  __syncthreads();

#pragma unroll
  for (int l = 0; l < 2; ++l) {
    const float* m0 = &mats[(l * 3 + 0) * 512];
    apply_q0(m0, m0 + 256, sr, si, lane, wid);
    __syncthreads();
    const float* m1 = &mats[(l * 3 + 1) * 512];
    apply_q1(m1, m1 + 256, sr, si, lane, wid);
    __syncthreads();
    const float* m2 = &mats[(l * 3 + 2) * 512];
    apply_q2(m2, m2 + 256, sr, si, lane, wid);
    __syncthreads();
    permute(sr, si, pm.m[l], tid);     // fused 12-CNOT ring (one gather/scatter)
  }

  // out[b] = head_b + sum_i |amp_i|^2 * sum_w head_w[w]*(1-2*bit(11-w))
  float hw[NW];
#pragma unroll
  for (int w = 0; w < NW; ++w) hw[w] = head_w[w];
  float local = 0.0f;
  for (int idx = tid; idx < DIM; idx += 256) {
    float p = sr[idx] * sr[idx] + si[idx] * si[idx];
    float wsum = 0.0f;
#pragma unroll
    for (int w = 0; w < NW; ++w)
      wsum += hw[w] * (1.0f - 2.0f * (float)((idx >> (11 - w)) & 1));
    local += p * wsum;
  }
  red[tid] = local;
  __syncthreads();
  for (int s = 128; s > 0; s >>= 1) {
    if (tid < s) red[tid] += red[tid + s];
    __syncthreads();
  }
  if (tid == 0) out[b] = red[0] + head_b[0];
}

extern "C" void kernel_launch(void* const* d_in, const int* in_sizes, int n_in,
                              void* d_out, int out_size, void* d_ws, size_t ws_size,
                              hipStream_t stream) {
  (void)in_sizes; (void)n_in; (void)out_size; (void)ws_size;
  const float* x      = (const float*)d_in[0];
  const float* var    = (const float*)d_in[1];
  const float* head_w = (const float*)d_in[2];
  const float* head_b = (const float*)d_in[3];
  float* out  = (float*)d_out;
  float* mats = (float*)d_ws;            // 6*512 floats = 12 KB scratch

  // Compose each layer's 12 CNOTs (applied in circuit order w=0..11,
  // ctrl wire w -> bit 11-w, tgt wire (w+r)%12 -> bit 11-((w+r)%12)) into one
  // GF(2)-linear source-index map: src(j) = XOR_{b set in j} colm[b].
  // s_final[i] = s_0[g_{w=0}(g_{w=1}(...g_{w=11}(i)))], so apply w=11 first.
  PermMasks pm;
  for (int l = 0; l < 2; ++l) {
    const int r = l + 1;
    for (int b = 0; b < NW; ++b) {
      unsigned v = 1u << b;
      for (int w = NW - 1; w >= 0; --w) {
        int pc = 11 - w;
        int pt = 11 - ((w + r) % NW);
        if ((v >> pc) & 1) v ^= (1u << pt);
      }
      pm.m[l][b] = (int)v;
    }
  }

  qnn_prep<<<6, 256, 0, stream>>>(var, mats);
  qnn_sim<<<BATCH, 256, 0, stream>>>(x, mats, head_w, head_b, out, pm);
}